// median_layer_37185826849046
// MI455X (gfx1250) — compile-verified
//
#include <hip/hip_runtime.h>
#include <hip/hip_bf16.h>
#include <stdint.h>

// ---------------------------------------------------------------------------
// 3x3 median filter (with the reference's 1-column scatter shift) + index gen.
// Memory-bound: ~265MB traffic -> ~11.4us at 23.3 TB/s. CDNA5 path used:
// async global->LDS staging (ASYNCcnt) + non-temporal 16B stores.
// ---------------------------------------------------------------------------

#define IMG_H 4096
#define IMG_W 4096
#define TILE  64
#define LDS_H 66            // rows r0-1 .. r0+64
#define LDS_W 68            // cols c0-4 .. c0+63  (17 x float4, 16B aligned)
#define CHUNKS_PER_ROW (LDS_W / 4)   // 17
#define NCHUNK (LDS_H * CHUNKS_PER_ROW) // 1122

typedef float v4f __attribute__((ext_vector_type(4)));
typedef int   v4i __attribute__((vector_size(16)));
typedef __attribute__((address_space(1))) v4i* glob_v4i_ptr;   // global int4*
typedef __attribute__((address_space(3))) v4i* lds_v4i_ptr;    // shared int4*

#if defined(__has_builtin)
#if __has_builtin(__builtin_amdgcn_global_load_async_to_lds_b128)
#define USE_ASYNC_LDS 1
#pragma message("CDNA5: async global_load_async_to_lds_b128 path ENABLED")
#endif
#endif
#ifndef USE_ASYNC_LDS
#pragma message("CDNA5: async-to-LDS builtin NOT found; direct LDS copy fallback")
#endif

__device__ __forceinline__ void mn_mx(float &a, float &b) {
    float mn = fminf(a, b);
    b = fmaxf(a, b);
    a = mn;
}

// Paeth 19-exchange median-of-9 network.
__device__ __forceinline__ float median9(float p0, float p1, float p2,
                                         float p3, float p4, float p5,
                                         float p6, float p7, float p8) {
    mn_mx(p1, p2); mn_mx(p4, p5); mn_mx(p7, p8);
    mn_mx(p0, p1); mn_mx(p3, p4); mn_mx(p6, p7);
    mn_mx(p1, p2); mn_mx(p4, p5); mn_mx(p7, p8);
    mn_mx(p0, p3); mn_mx(p5, p8); mn_mx(p4, p7);
    mn_mx(p3, p6); mn_mx(p1, p4); mn_mx(p2, p5);
    mn_mx(p4, p7); mn_mx(p4, p2); mn_mx(p6, p4);
    mn_mx(p4, p2);
    return p4;
}

// One block computes a 64x64 output tile. Input region (66 rows x 68 cols)
// is staged into LDS with per-lane 16B async copies, then each thread
// produces 4 consecutive output pixels per pass (float4 NT store).
__global__ __launch_bounds__(256)
void median_scatter_kernel(const float* __restrict__ in, float* __restrict__ out) {
    __shared__ float lds[LDS_H * LDS_W];

    const int t  = threadIdx.x;
    const int c0 = blockIdx.x * TILE;
    const int r0 = blockIdx.y * TILE;

    // ---- Stage input region: rows [r0-1, r0+64], cols [c0-4, c0+63] ----
    for (int q = t; q < NCHUNK; q += 256) {
        const int row = q / CHUNKS_PER_ROW;
        const int ch  = q - row * CHUNKS_PER_ROW;
        int gr = r0 - 1 + row;
        gr = gr < 0 ? 0 : (gr > IMG_H - 1 ? IMG_H - 1 : gr);
        int gc = c0 - 4 + ch * 4;        // 16B aligned
        gc = gc < 0 ? 0 : gc;            // clamped halo is never consumed
        const float* gp = in + (size_t)gr * IMG_W + gc;
        float* lp = &lds[row * LDS_W + ch * 4];
#ifdef USE_ASYNC_LDS
        __builtin_amdgcn_global_load_async_to_lds_b128(
            (glob_v4i_ptr)(uintptr_t)gp,
            (lds_v4i_ptr)(uint32_t)(uintptr_t)lp,
            0, 0);
#else
        *(v4f*)lp = *(const v4f*)gp;
#endif
    }
#ifdef USE_ASYNC_LDS
#if defined(__has_builtin) && __has_builtin(__builtin_amdgcn_s_wait_asynccnt)
    __builtin_amdgcn_s_wait_asynccnt(0);
#else
    asm volatile("s_wait_asynccnt 0" ::: "memory");
#endif
#endif
    __syncthreads();

    // ---- Compute: thread -> 4 consecutive cols, 4 row-passes of 16 rows ----
    const int tx4 = (t & 15) * 4;        // 0..60
    const int ty  = t >> 4;              // 0..15
    #pragma unroll
    for (int p = 0; p < 4; ++p) {
        const int r = r0 + p * 16 + ty;
        const int L = r - r0 + 1;        // LDS row of global row r
        float res[4];
        #pragma unroll
        for (int e = 0; e < 4; ++e) {
            const int c = c0 + tx4 + e;
            const int X = c - c0 + 4;    // LDS col of global col c
            if (r >= 1 && r <= IMG_H - 2 && c >= 2) {
                // median centered at (r, c-1): rows L-1..L+1, cols X-2..X
                const float* a = &lds[(L - 1) * LDS_W + X - 2];
                const float* b = &lds[(L    ) * LDS_W + X - 2];
                const float* d = &lds[(L + 1) * LDS_W + X - 2];
                res[e] = median9(a[0], a[1], a[2],
                                 b[0], b[1], b[2],
                                 d[0], d[1], d[2]);
            } else {
                res[e] = lds[L * LDS_W + X];   // passthrough border
            }
        }
        v4f o = { res[0], res[1], res[2], res[3] };
        __builtin_nontemporal_store(o, (v4f*)(out + (size_t)r * IMG_W + c0 + tx4));
    }
}

// Index pairs: pair k = (1 + k/4094, (k%4094) + 2), written as fp32 (exact,
// values < 2^24). Two pairs per thread => one coalesced float4 NT store.
__global__ __launch_bounds__(256)
void coord_indices_kernel(float* __restrict__ out) {
    const int NPW = IMG_W - 2;                    // 4094
    const int NP  = (IMG_H - 2) * (IMG_W - 2);    // 16760836 (even)
    const long long tid = (long long)blockIdx.x * 256 + threadIdx.x;
    if (tid >= NP / 2) return;
    const int k = (int)(tid << 1);
    const int i  = k / NPW;
    const int j  = k - i * NPW;
    int i2 = i, j2 = j + 1;
    if (j2 == NPW) { j2 = 0; i2 = i + 1; }
    v4f o = { (float)(i + 1), (float)(j + 2),
              (float)(i2 + 1), (float)(j2 + 2) };
    float* base = out + (size_t)IMG_H * IMG_W;
    __builtin_nontemporal_store(o, (v4f*)(base + ((size_t)tid << 2)));
}

extern "C" void kernel_launch(void* const* d_in, const int* in_sizes, int n_in,
                              void* d_out, int out_size, void* d_ws, size_t ws_size,
                              hipStream_t stream) {
    (void)in_sizes; (void)n_in; (void)out_size; (void)d_ws; (void)ws_size;
    const float* in = (const float*)d_in[0];
    float* out = (float*)d_out;

    dim3 grid(IMG_W / TILE, IMG_H / TILE);   // 64 x 64 tiles
    median_scatter_kernel<<<grid, 256, 0, stream>>>(in, out);

    const int np2 = ((IMG_H - 2) * (IMG_W - 2)) / 2;   // float4 stores
    const int blocks = (np2 + 255) / 256;
    coord_indices_kernel<<<blocks, 256, 0, stream>>>(out);
}